// WordEmbeddings_58849641890511
// MI455X (gfx1250) — compile-verified
//
#include <hip/hip_runtime.h>

#define VOC_SIZE  1000000
#define EMBED_DIM 64
#define N_UPD     262144
#define N_Q       (4096 * 200)

typedef __attribute__((ext_vector_type(4))) float v4f;

// ---------------------------------------------------------------------------
// Kernel 1: copy embedding table -> workspace via CDNA5 async LDS DMA path.
// Each lane moves 16 bytes: global -> LDS (async) -> global (async), tracked
// with ASYNCcnt. This bypasses VGPRs entirely (tensor/async data path).
// ---------------------------------------------------------------------------
__global__ __launch_bounds__(256) void we_copy_async(const float* __restrict__ src,
                                                     float* __restrict__ dst,
                                                     int n4) {
  __shared__ char stage[256 * 16];
  int tid = blockIdx.x * 256 + threadIdx.x;
  if (tid < n4) {
    // Low 32 bits of the generic pointer to LDS == workgroup-relative LDS byte
    // address (ISA 10.2).
    unsigned lds_off = (unsigned)(unsigned long long)(&stage[threadIdx.x * 16]);
    unsigned long long g_src = (unsigned long long)src + (unsigned long long)tid * 16ull;
    unsigned long long g_dst = (unsigned long long)dst + (unsigned long long)tid * 16ull;
    asm volatile(
        "global_load_async_to_lds_b128 %0, %1, off\n\t"
        "s_wait_asynccnt 0\n\t"
        "global_store_async_from_lds_b128 %2, %0, off\n\t"
        "s_wait_asynccnt 0"
        :
        : "v"(lds_off), "v"(g_src), "v"(g_dst)
        : "memory");
  }
}

// ---------------------------------------------------------------------------
// Kernel 2: scatter-add updates into the workspace table with hardware fp32
// global atomics (global_atomic_add_f32, no CAS loop). One thread per element;
// 64 consecutive floats per row -> fully coalesced 256B atomic bursts.
// ---------------------------------------------------------------------------
__global__ __launch_bounds__(256) void we_scatter(const int* __restrict__ idx,
                                                  const float* __restrict__ upd,
                                                  float* __restrict__ table,
                                                  int n) {
  int t = blockIdx.x * 256 + threadIdx.x;
  if (t < n) {
    int row = t >> 6;        // / EMBED_DIM
    int col = t & 63;        // % EMBED_DIM
    long long d = (long long)idx[row];
    unsafeAtomicAdd(&table[d * EMBED_DIM + col], upd[t]);
  }
}

// ---------------------------------------------------------------------------
// Kernel 3: gather rows of the updated table. One float4 per thread (16 lanes
// cover one 256B row). Table reads are RT (exploit 192MB L2); output is a
// 210MB write-once stream -> non-temporal 128-bit stores.
// ---------------------------------------------------------------------------
__global__ __launch_bounds__(256) void we_gather(const v4f* __restrict__ table,
                                                 const int* __restrict__ qs,
                                                 v4f* __restrict__ out,
                                                 int n4) {
  int t = blockIdx.x * 256 + threadIdx.x;
  if (t < n4) {
    int row = t >> 4;        // output row
    int c   = t & 15;        // float4 column within row
    long long q = (long long)qs[row];
    v4f v = table[q * (EMBED_DIM / 4) + c];
    __builtin_nontemporal_store(v, &out[t]);
  }
}

extern "C" void kernel_launch(void* const* d_in, const int* in_sizes, int n_in,
                              void* d_out, int out_size, void* d_ws, size_t ws_size,
                              hipStream_t stream) {
  const float* kern    = (const float*)d_in[0];  // (VOC_SIZE, 64) fp32
  const int*   indices = (const int*)d_in[1];    // (N_UPD,)
  const float* upd     = (const float*)d_in[2];  // (N_UPD, 64) fp32
  const int*   qs      = (const int*)d_in[3];    // (B, L)
  float*       out     = (float*)d_out;          // (B, L, 64) fp32
  float*       table   = (float*)d_ws;           // scratch copy of the table (256 MB)

  // 1) table -> workspace (async DMA copy), 16M float4
  int n4_tab = VOC_SIZE * EMBED_DIM / 4;
  we_copy_async<<<(n4_tab + 255) / 256, 256, 0, stream>>>(kern, table, n4_tab);

  // 2) scatter-add updates (atomic fp32)
  int n_sc = N_UPD * EMBED_DIM;
  we_scatter<<<(n_sc + 255) / 256, 256, 0, stream>>>(indices, upd, table, n_sc);

  // 3) gather query rows, NT stores
  int n4_out = N_Q * (EMBED_DIM / 4);
  we_gather<<<(n4_out + 255) / 256, 256, 0, stream>>>((const v4f*)table, qs,
                                                      (v4f*)out, n4_out);
}